// EGCNGuard_9363028706301
// MI455X (gfx1250) — compile-verified
//
#include <hip/hip_runtime.h>
#include <math.h>

#define D 128
#define THRESHOLD 0.1f
#define LN_EPS 1e-5f
#define COS_EPS 1e-8f

typedef __attribute__((ext_vector_type(2))) float v2f;
typedef __attribute__((ext_vector_type(8))) float v8f;

// ---------------- wave32 butterfly reductions ----------------
__device__ __forceinline__ float wred_sum(float v) {
  for (int o = 16; o > 0; o >>= 1) v += __shfl_xor(v, o, 32);
  return v;
}
__device__ __forceinline__ float wred_max(float v) {
  for (int o = 16; o > 0; o >>= 1) v = fmaxf(v, __shfl_xor(v, o, 32));
  return v;
}

// ---------------- per-node: feature norms + zero deg/cnt ----------------
__global__ void node_prep(const float* __restrict__ F, float* __restrict__ norms,
                          float* __restrict__ deg, float* __restrict__ cnt, int N) {
  int wid = (blockIdx.x * blockDim.x + threadIdx.x) >> 5;
  int lane = threadIdx.x & 31;
  if (wid >= N) return;
  float4 f = reinterpret_cast<const float4*>(F + (size_t)wid * D)[lane];
  float p = wred_sum(f.x * f.x + f.y * f.y + f.z * f.z + f.w * f.w);
  if (lane == 0) {
    norms[wid] = sqrtf(p);
    deg[wid] = 0.f;
    cnt[wid] = 0.f;
  }
}

// ---------------- dense GEMM G = F(N x 128) @ W(128 x 128) via f32 WMMA ----------------
// One wave computes a 16x128 row strip: 8 accumulator tiles share each A fragment,
// so the strided A loads are issued once per K-step and reused by 8 WMMAs.
__global__ void gemm128_wmma(const float* __restrict__ F, const float* __restrict__ W,
                             float* __restrict__ G, int nrows) {
  int wid = (blockIdx.x * blockDim.x + threadIdx.x) >> 5;  // one wave per 16-row strip
  int lane = threadIdx.x & 31;
  if (wid * 16 >= nrows) return;
  int mr = wid * 16 + (lane & 15);         // A row this lane feeds
  int nl = lane & 15;                      // B/C column-within-tile this lane feeds
  int kh = (lane >> 4) << 1;               // K sub-offset: 0 (lanes 0-15) or 2 (lanes 16-31)

  v8f acc[8];
#pragma unroll
  for (int t = 0; t < 8; ++t) acc[t] = (v8f){};

  const float* fp = F + (size_t)mr * D + kh;
  const float* wp = W + (size_t)kh * D + nl;
  for (int k = 0; k < D; k += 4) {
    v2f a;
    a.x = fp[0];                           // A[mr, k+kh]
    a.y = fp[1];                           // A[mr, k+kh+1]
#pragma unroll
    for (int t = 0; t < 8; ++t) {
      v2f b;
      b.x = wp[t * 16];                    // B[k+kh,   t*16+nl]
      b.y = wp[t * 16 + D];                // B[k+kh+1, t*16+nl]
      acc[t] = __builtin_amdgcn_wmma_f32_16x16x4_f32(false, a, false, b, (short)0,
                                                     acc[t], false, false);
    }
    fp += 4;
    wp += 4 * D;
  }
  // C/D layout: VGPR j -> M = j (lanes 0-15) / M = 8+j (lanes 16-31), N = lane&15
  int r0 = wid * 16 + ((lane >> 4) << 3);
#pragma unroll
  for (int t = 0; t < 8; ++t) {
    float* gp = G + (size_t)r0 * D + t * 16 + nl;
#pragma unroll
    for (int j = 0; j < 8; ++j)
      gp[(size_t)j * D] = acc[t][j];
  }
}

// ---------------- edge pass 1: cosine sim, mask, deg/cnt segment sums ----------------
__global__ void edge_pass1(const float* __restrict__ F, const float* __restrict__ norms,
                           const int* __restrict__ row, const int* __restrict__ col,
                           int* __restrict__ alive, float* __restrict__ s_eff,
                           float* __restrict__ deg, float* __restrict__ cnt,
                           int E, int first) {
  int wid = (blockIdx.x * blockDim.x + threadIdx.x) >> 5;
  int lane = threadIdx.x & 31;
  if (wid >= E) return;
  int r = row[wid], c = col[wid];
  float4 fr = reinterpret_cast<const float4*>(F + (size_t)r * D)[lane];
  float4 fc = reinterpret_cast<const float4*>(F + (size_t)c * D)[lane];
  float p = wred_sum(fr.x * fc.x + fr.y * fc.y + fr.z * fc.z + fr.w * fc.w);
  if (lane == 0) {
    float nr = fmaxf(norms[r], COS_EPS);
    float ncv = fmaxf(norms[c], COS_EPS);
    float sims = p / (nr * ncv);
    int a = first ? 1 : alive[wid];
    int mask = (a && (sims >= THRESHOLD)) ? 1 : 0;
    float se = mask ? sims : 0.f;
    alive[wid] = mask;       // next layer's alive (mask | is_self; is_self==false here)
    s_eff[wid] = se;
    if (se != 0.f) atomicAdd(&deg[r], se);
    if (mask) atomicAdd(&cnt[c], 1.f);
  }
}

// ---------------- per-node: dinv, self-loop value, init H = b + selfv*G[i] ----------------
__global__ void node_mid(const float* __restrict__ G, const float* __restrict__ bias,
                         const float* __restrict__ deg, const float* __restrict__ cnt,
                         float* __restrict__ dinv, float* __restrict__ H,
                         int N, float selfadd) {
  int wid = (blockIdx.x * blockDim.x + threadIdx.x) >> 5;
  int lane = threadIdx.x & 31;
  if (wid >= N) return;
  float d = deg[wid];
  float dv = (d > 0.f) ? rsqrtf(d) : 0.f;
  float sv = expf(1.f / (cnt[wid] + selfadd + 1.f));   // exp(dinv_new)
  if (lane == 0) dinv[wid] = dv;
  float4 g = reinterpret_cast<const float4*>(G + (size_t)wid * D)[lane];
  float4 b = reinterpret_cast<const float4*>(bias)[lane];
  float4 h;
  h.x = b.x + sv * g.x;
  h.y = b.y + sv * g.y;
  h.z = b.z + sv * g.z;
  h.w = b.w + sv * g.w;
  reinterpret_cast<float4*>(H + (size_t)wid * D)[lane] = h;
}

// ---------------- edge pass 2: val = exp(dinv_r*s_eff*dinv_c); H[r] += val * G[c] ----------------
__global__ void edge_pass2(const float* __restrict__ G,
                           const int* __restrict__ row, const int* __restrict__ col,
                           const int* __restrict__ alive, const float* __restrict__ s_eff,
                           const float* __restrict__ dinv, float* __restrict__ H, int E) {
  int wid = (blockIdx.x * blockDim.x + threadIdx.x) >> 5;
  int lane = threadIdx.x & 31;
  if (wid >= E) return;
  if (!alive[wid]) return;   // wave-uniform
  int r = row[wid], c = col[wid];
  float val = expf(dinv[r] * s_eff[wid] * dinv[c]);
  float4 g = reinterpret_cast<const float4*>(G + (size_t)c * D)[lane];
  float* hp = H + (size_t)r * D + lane * 4;
  atomicAdd(hp + 0, val * g.x);
  atomicAdd(hp + 1, val * g.y);
  atomicAdd(hp + 2, val * g.z);
  atomicAdd(hp + 3, val * g.w);
}

// ---------------- per-node: LayerNorm + ReLU (in place) ----------------
__global__ void ln_relu(float* __restrict__ H, const float* __restrict__ g,
                        const float* __restrict__ b, int N) {
  int wid = (blockIdx.x * blockDim.x + threadIdx.x) >> 5;
  int lane = threadIdx.x & 31;
  if (wid >= N) return;
  float4* hp = reinterpret_cast<float4*>(H + (size_t)wid * D) + lane;
  float4 h = *hp;
  float mu = wred_sum(h.x + h.y + h.z + h.w) * (1.f / D);
  float dx = h.x - mu, dy = h.y - mu, dz = h.z - mu, dw = h.w - mu;
  float var = wred_sum(dx * dx + dy * dy + dz * dz + dw * dw) * (1.f / D);
  float rs = rsqrtf(var + LN_EPS);
  float4 g4 = reinterpret_cast<const float4*>(g)[lane];
  float4 b4 = reinterpret_cast<const float4*>(b)[lane];
  h.x = fmaxf(dx * rs * g4.x + b4.x, 0.f);
  h.y = fmaxf(dy * rs * g4.y + b4.y, 0.f);
  h.z = fmaxf(dz * rs * g4.z + b4.z, 0.f);
  h.w = fmaxf(dw * rs * g4.w + b4.w, 0.f);
  *hp = h;
}

// ---------------- per-node: log-softmax -> d_out ----------------
__global__ void log_softmax_k(const float* __restrict__ H, float* __restrict__ out, int N) {
  int wid = (blockIdx.x * blockDim.x + threadIdx.x) >> 5;
  int lane = threadIdx.x & 31;
  if (wid >= N) return;
  float4 h = reinterpret_cast<const float4*>(H + (size_t)wid * D)[lane];
  float m = wred_max(fmaxf(fmaxf(h.x, h.y), fmaxf(h.z, h.w)));
  float s = wred_sum(expf(h.x - m) + expf(h.y - m) + expf(h.z - m) + expf(h.w - m));
  float l = m + logf(s);
  float4 o;
  o.x = h.x - l;
  o.y = h.y - l;
  o.z = h.z - l;
  o.w = h.w - l;
  reinterpret_cast<float4*>(out + (size_t)wid * D)[lane] = o;
}

extern "C" void kernel_launch(void* const* d_in, const int* in_sizes, int n_in,
                              void* d_out, int out_size, void* d_ws, size_t ws_size,
                              hipStream_t stream) {
  (void)n_in; (void)out_size; (void)ws_size;
  const float* x    = (const float*)d_in[0];
  const int*   row  = (const int*)d_in[1];
  const int*   col  = (const int*)d_in[2];
  const float* W0   = (const float*)d_in[3];
  const float* b0   = (const float*)d_in[4];
  const float* W1   = (const float*)d_in[5];
  const float* b1   = (const float*)d_in[6];
  const float* ln1g = (const float*)d_in[7];
  const float* ln1b = (const float*)d_in[8];
  const float* ln2g = (const float*)d_in[9];
  const float* ln2b = (const float*)d_in[10];

  const int N = in_sizes[0] / D;
  const int E = in_sizes[1];

  // workspace carve-out (all 512B aligned)
  char* ws = (char*)d_ws;
  size_t off = 0;
  auto carve = [&](size_t bytes) {
    void* p = ws + off;
    off = (off + bytes + 511) & ~(size_t)511;
    return p;
  };
  float* A     = (float*)carve((size_t)N * D * sizeof(float));  // feature/output buffer
  float* G     = (float*)carve((size_t)N * D * sizeof(float));  // F @ W result
  float* norms = (float*)carve((size_t)N * sizeof(float));
  float* deg   = (float*)carve((size_t)N * sizeof(float));
  float* cnt   = (float*)carve((size_t)N * sizeof(float));
  float* dinv  = (float*)carve((size_t)N * sizeof(float));
  float* s_eff = (float*)carve((size_t)E * sizeof(float));
  int*   alive = (int*)carve((size_t)E * sizeof(int));

  const int TB = 256;                              // 8 waves per block
  const int nodeBlocks = (N + 7) / 8;              // wave per node
  const int edgeBlocks = (E + 7) / 8;              // wave per edge
  const int gemmWaves  = (N + 15) / 16;            // wave per 16-row strip
  const int gemmBlocks = (gemmWaves + 7) / 8;

  auto run_layer = [&](const float* F, const float* Wt, const float* bias,
                       int first, float selfadd) {
    node_prep<<<nodeBlocks, TB, 0, stream>>>(F, norms, deg, cnt, N);
    gemm128_wmma<<<gemmBlocks, TB, 0, stream>>>(F, Wt, G, N);
    edge_pass1<<<edgeBlocks, TB, 0, stream>>>(F, norms, row, col, alive, s_eff,
                                              deg, cnt, E, first);
    node_mid<<<nodeBlocks, TB, 0, stream>>>(G, bias, deg, cnt, dinv, A, N, selfadd);
    edge_pass2<<<edgeBlocks, TB, 0, stream>>>(G, row, col, alive, s_eff, dinv, A, E);
  };

  // layer 1: conv0(x) -> LN1 -> ReLU    (self-loops not yet counted in cnt)
  run_layer(x, W0, b0, /*first=*/1, /*selfadd=*/0.f);
  ln_relu<<<nodeBlocks, TB, 0, stream>>>(A, ln1g, ln1b, N);
  // layer 2: conv1(h) -> LN2 -> ReLU    (self-loops alive => cnt += 1)
  run_layer(A, W1, b1, /*first=*/0, /*selfadd=*/1.f);
  ln_relu<<<nodeBlocks, TB, 0, stream>>>(A, ln2g, ln2b, N);
  // layer 3: reference bug reuses (W1, b1); then log-softmax
  run_layer(A, W1, b1, /*first=*/0, /*selfadd=*/1.f);
  log_softmax_k<<<nodeBlocks, TB, 0, stream>>>(A, (float*)d_out, N);
}